// fTN_NN_Model_1391569404361
// MI455X (gfx1250) — compile-verified
//
#include <hip/hip_runtime.h>

#define NN_ETA 0.001f

typedef __attribute__((ext_vector_type(2))) float v2f;
typedef __attribute__((ext_vector_type(8))) float v8f;

// A layout: (LX=8, LY=4, D=4, D, D, D, PHYS=2) row-major
__device__ __forceinline__ int aidx(int x, int y, int i0, int i1, int i2, int i3, int p) {
    return ((((((x * 4 + y) * 4 + i0) * 4 + i1) * 4 + i2) * 4 + i3) * 2 + p);
}

// ---------------------------------------------------------------------------
// Kernel 1: h[b][j] = relu(cfg[b,:] . W1[:,j] + b1[j]);  grid=128, block=64
// ---------------------------------------------------------------------------
__global__ __launch_bounds__(64) void h_kernel(const int* __restrict__ cfg,
                                               const float* __restrict__ W1,
                                               const float* __restrict__ b1,
                                               float* __restrict__ h) {
    int b = blockIdx.x;
    int j = threadIdx.x;
    float acc = b1[j];
    for (int k = 0; k < 32; ++k)
        acc += (float)cfg[b * 32 + k] * W1[k * 64 + j];
    h[b * 64 + j] = fmaxf(acc, 0.0f);
}

// ---------------------------------------------------------------------------
// Kernel 2: corr(128x16384) = NN_ETA * (h(128x64) @ W2(64x16384) + b2)
// fp32 WMMA 16x16x4; one wave per 16x16 output tile, 8 waves/block.
// grid = 1024 column-tiles, block = 256 (wave w -> row-tile w, 8*16=128 rows)
// ---------------------------------------------------------------------------
__global__ __launch_bounds__(256) void corr_wmma_kernel(const float* __restrict__ h,
                                                        const float* __restrict__ W2,
                                                        const float* __restrict__ b2,
                                                        float* __restrict__ corr) {
#if __has_builtin(__builtin_amdgcn_wmma_f32_16x16x4_f32)
    int lane  = threadIdx.x & 31;
    int rt    = threadIdx.x >> 5;   // row tile 0..7
    int ct    = blockIdx.x;         // col tile 0..1023
    int m15   = lane & 15;
    int khalf = lane >> 4;          // 0 or 1
    int colbase = ct * 16;
    int n = colbase + m15;

    v8f acc = {};
    for (int kc = 0; kc < 16; ++kc) {
        int k0 = kc * 4 + khalf * 2;
        v2f a, bfr;
        // A frag (16x4): lanes 0-15 hold K=k0,k0+1 ; lanes 16-31 K=k0+2,k0+3
        a.x   = h[(rt * 16 + m15) * 64 + k0];
        a.y   = h[(rt * 16 + m15) * 64 + k0 + 1];
        // B frag (4x16): row K striped over lanes (N = lane&15)
        bfr.x = W2[(size_t)k0 * 16384 + n];
        bfr.y = W2[(size_t)(k0 + 1) * 16384 + n];
        acc = __builtin_amdgcn_wmma_f32_16x16x4_f32(
            /*neg_a=*/false, a, /*neg_b=*/false, bfr,
            /*c_mod=*/(short)0, acc, /*reuse_a=*/false, /*reuse_b=*/false);
    }
    // D layout: VGPR r -> M = r (lanes 0-15) / r+8 (lanes 16-31), N = lane&15
    int mbase = rt * 16 + (khalf ? 8 : 0);
    float bb = b2[n];
    for (int r = 0; r < 8; ++r)
        corr[(size_t)(mbase + r) * 16384 + n] = NN_ETA * (acc[r] + bb);
#else
    // Scalar fallback (should not trigger on gfx1250)
    size_t gid = (size_t)blockIdx.x * blockDim.x + threadIdx.x;
    for (size_t e = gid; e < (size_t)128 * 16384; e += (size_t)gridDim.x * blockDim.x) {
        int m = (int)(e >> 14), n = (int)(e & 16383);
        float acc = b2[n];
        for (int k = 0; k < 64; ++k) acc += h[m * 64 + k] * W2[(size_t)k * 16384 + n];
        corr[e] = NN_ETA * acc;
    }
#endif
}

// ---------------------------------------------------------------------------
// Kernel 3: per-config amplitude. One 256-thread workgroup per config.
// Builds bot2/top2 per site-column x in LDS, applies NN correction, runs the
// factored transfer-matrix chain  T[I,j,u]=Σ_i v[iI]·bot2 ; v'[jJ]=Σ_{I,u} T·top2
// If corr==nullptr, computes h and the correction inline (workspace fallback).
// ---------------------------------------------------------------------------
__global__ __launch_bounds__(256) void amp_kernel(const int* __restrict__ cfg,
                                                  const float* __restrict__ A,
                                                  const float* __restrict__ W1,
                                                  const float* __restrict__ b1,
                                                  const float* __restrict__ W2,
                                                  const float* __restrict__ b2,
                                                  const float* __restrict__ corr,
                                                  float* __restrict__ out) {
    __shared__ int   cfgS[32];
    __shared__ float hS[64];
    __shared__ float r0s[64], r1s[256], r2s[256], r3s[64];
    __shared__ float sbot[1024], stopv[1024], Ts[1024];
    __shared__ float v[256], vn[256];

    int b = blockIdx.x;
    int tid = threadIdx.x;

    if (tid < 32) cfgS[tid] = cfg[b * 32 + tid];
    v[tid] = (tid == 0) ? 1.0f : 0.0f;
    __syncthreads();

    bool haveCorr = (corr != nullptr);
    if (!haveCorr && tid < 64) {
        float acc = b1[tid];
        for (int k = 0; k < 32; ++k) acc += (float)cfgS[k] * W1[k * 64 + tid];
        hS[tid] = fmaxf(acc, 0.0f);
    }
    __syncthreads();

    for (int x = 0; x < 8; ++x) {
        int c0 = cfgS[x * 4 + 0], c1 = cfgS[x * 4 + 1];
        int c2 = cfgS[x * 4 + 2], c3 = cfgS[x * 4 + 3];

        // Stage selected site tensors for this column x.
        if (tid < 64) {
            int l = tid >> 4, r = (tid >> 2) & 3, m = tid & 3;
            r0s[tid] = A[aidx(x, 0, l, r, 0, m, c0)];  // s[x,0,l,r,0,m]
            r3s[tid] = A[aidx(x, 3, l, r, m, 0, c3)];  // s[x,3,L,R,m,0]
        }
        {
            int a0 = (tid >> 6) & 3, a1 = (tid >> 4) & 3, a2 = (tid >> 2) & 3, a3 = tid & 3;
            r1s[tid] = A[aidx(x, 1, a0, a1, a2, a3, c1)]; // [L][R][m][U]
            r2s[tid] = A[aidx(x, 2, a0, a1, a2, a3, c2)]; // [l][r][d][m]
        }
        __syncthreads();

        // bot2[x,i,j,u] / top2[x,i,j,u] with i=(l,L), j=(r,R)  (+ NN correction)
        for (int t = tid; t < 1024; t += 256) {
            int i = t >> 6, j = (t >> 2) & 15, u = t & 3;
            int l = i >> 2, L = i & 3, r = j >> 2, R = j & 3;
            float ab = 0.0f, at = 0.0f;
            for (int m = 0; m < 4; ++m) {
                ab += r0s[l * 16 + r * 4 + m] * r1s[((L * 4 + R) * 4 + m) * 4 + u];
                at += r2s[((l * 4 + r) * 4 + u) * 4 + m] * r3s[L * 16 + R * 4 + m];
            }
            int nb = x * 1024 + t;  // flat index inside the 8192-long halves
            float cb, ctp;
            if (haveCorr) {
                cb  = corr[(size_t)b * 16384 + nb];
                ctp = corr[(size_t)b * 16384 + 8192 + nb];
            } else {
                float sb = b2[nb], st = b2[8192 + nb];
                for (int k = 0; k < 64; ++k) {
                    sb += hS[k] * W2[(size_t)k * 16384 + nb];
                    st += hS[k] * W2[(size_t)k * 16384 + 8192 + nb];
                }
                cb = NN_ETA * sb; ctp = NN_ETA * st;
            }
            sbot[t]  = ab + cb;
            stopv[t] = at + ctp;
        }
        __syncthreads();

        // T[I,j,u] = Σ_i v[i*16+I] * bot2[i,j,u]
        for (int t = tid; t < 1024; t += 256) {
            int I = t >> 6, ju = t & 63;
            float acc = 0.0f;
            for (int i = 0; i < 16; ++i) acc += v[i * 16 + I] * sbot[i * 64 + ju];
            Ts[t] = acc;
        }
        __syncthreads();

        // v'[j*16+J] = Σ_{I,u} T[I,j,u] * top2[I,J,u]
        {
            int j = tid >> 4, J = tid & 15;
            float acc = 0.0f;
            for (int I = 0; I < 16; ++I)
                for (int u = 0; u < 4; ++u)
                    acc += Ts[I * 64 + j * 4 + u] * stopv[I * 64 + J * 4 + u];
            vn[tid] = acc;
        }
        __syncthreads();
        v[tid] = vn[tid];
        __syncthreads();
    }

    if (tid == 0) out[b] = v[0];
}

// ---------------------------------------------------------------------------
extern "C" void kernel_launch(void* const* d_in, const int* in_sizes, int n_in,
                              void* d_out, int out_size, void* d_ws, size_t ws_size,
                              hipStream_t stream) {
    const int*   cfg = (const int*)d_in[0];
    const float* A   = (const float*)d_in[1];
    const float* W1  = (const float*)d_in[2];
    const float* b1  = (const float*)d_in[3];
    const float* W2  = (const float*)d_in[4];
    const float* b2  = (const float*)d_in[5];
    float* out = (float*)d_out;

    const size_t needH = (size_t)128 * 64 * sizeof(float);       // 32 KB
    const size_t needC = (size_t)128 * 16384 * sizeof(float);    // 8 MB

    if (ws_size >= needH + needC) {
        float* h    = (float*)d_ws;
        float* corr = (float*)((char*)d_ws + needH);
        h_kernel<<<128, 64, 0, stream>>>(cfg, W1, b1, h);
        corr_wmma_kernel<<<1024, 256, 0, stream>>>(h, W2, b2, corr);
        amp_kernel<<<128, 256, 0, stream>>>(cfg, A, W1, b1, W2, b2, corr, out);
    } else {
        amp_kernel<<<128, 256, 0, stream>>>(cfg, A, W1, b1, W2, b2, nullptr, out);
    }
}